// LTCCell_45595372815185
// MI455X (gfx1250) — compile-verified
//
#include <hip/hip_runtime.h>
#include <stdint.h>

// LTC cell for MI455X (gfx1250, wave32).
// B=1024, I=128, N=256, 6 unfolds. Compute-bound on VALU/trans pipe;
// params are L2-resident. WMMA is inapplicable (nonlinear coupling of j,n).
// CDNA5-specific path: global_load_async_to_lds_b128 + s_wait_asynccnt.

#define B_TOT   1024
#define I_DIM   128
#define N_DIM   256
#define UNFOLDS 6
#define BB      4     // batches per block (occupancy vs. L2-traffic tradeoff)
#define BN      128   // n-columns per block
#define NTHREADS 128  // == BN, 4 waves of 32

#ifndef __has_builtin
#define __has_builtin(x) 0
#endif

#if __has_builtin(__builtin_amdgcn_global_load_async_to_lds_b128)
#define HAVE_ASYNC_LDS 1
#else
#define HAVE_ASYNC_LDS 0
#endif
#if __has_builtin(__builtin_amdgcn_s_wait_asynccnt)
#define HAVE_WAIT_ASYNC 1
#else
#define HAVE_WAIT_ASYNC 0
#endif

// 128-bit payload type the async-LDS builtin expects (AS(1)/AS(3) int4*).
typedef int v4i __attribute__((vector_size(16)));
typedef __attribute__((address_space(1))) v4i gbl_v4i;
typedef __attribute__((address_space(3))) v4i lds_v4i;

__device__ __forceinline__ gbl_v4i* to_gbl(const void* p) {
  return (gbl_v4i*)(uintptr_t)p;      // flat AS(1) value == generic value
}
__device__ __forceinline__ lds_v4i* to_lds(void* p) {
  // flat->LDS: low 32 bits of a flat LDS address are the LDS byte offset
  return (lds_v4i*)(uint32_t)(uintptr_t)p;
}

__device__ __forceinline__ void async_copy16(void* lds_dst, const float* g_src) {
#if HAVE_ASYNC_LDS
  __builtin_amdgcn_global_load_async_to_lds_b128(to_gbl(g_src), to_lds(lds_dst), 0, 0);
#else
  *(float4*)lds_dst = *(const float4*)g_src;
#endif
}

__device__ __forceinline__ void wait_async0() {
#if HAVE_ASYNC_LDS
#if HAVE_WAIT_ASYNC
  __builtin_amdgcn_s_wait_asynccnt(0);
#else
  asm volatile("s_wait_asynccnt 0" ::: "memory");
#endif
#endif
}

__device__ __forceinline__ float fexp2(float x) { return __builtin_amdgcn_exp2f(x); }
__device__ __forceinline__ float frcp(float x)  { return __builtin_amdgcn_rcpf(x); }

#define L2E 1.44269504088896340736f

// ---------------------------------------------------------------------------
// Pack kernel: P[idx] = { mu*sigma*log2e, -sigma*log2e, W*erev, W }
// Removes 3 redundant VALU muls per (j,n) per block-group and fuses the four
// param streams into one coalesced b128 load in the hot loops.
// ---------------------------------------------------------------------------
__global__ __launch_bounds__(256) void ltc_pack(
    const float* __restrict__ mu, const float* __restrict__ sigma,
    const float* __restrict__ W, const float* __restrict__ erev,
    float4* __restrict__ packed, int total)
{
  const int idx = blockIdx.x * 256 + threadIdx.x;
  if (idx < total) {
    const float sl = sigma[idx] * L2E;
    const float w  = W[idx];
    packed[idx] = make_float4(mu[idx] * sl, -sl, w * erev[idx], w);
  }
}

// ---------------------------------------------------------------------------
// Kernel 1: sensory synapses -> w_num_sensory[b,n], w_den_sensory[b,n]
// ---------------------------------------------------------------------------
__global__ __launch_bounds__(NTHREADS) void ltc_sensory(
    const float* __restrict__ inputs, const float* __restrict__ input_w,
    const float* __restrict__ input_b,
    const float4* __restrict__ psens,     // packed [I,N]
    float* __restrict__ wnum_s, float* __restrict__ wden_s)
{
  __shared__ __align__(16) float xs[BB * I_DIM];   // 2 KB
  const int b0 = (blockIdx.x >> 1) * BB;
  const int n0 = (blockIdx.x & 1) * BN;
  const int t  = threadIdx.x;

  // async-stage the BB x 128 input tile into LDS (compile-time trip count)
  constexpr int XCOPIES = (BB * I_DIM / 4) / NTHREADS;   // = 1
#pragma unroll
  for (int k = 0; k < XCOPIES; ++k) {
    const int v = t + k * NTHREADS;
    async_copy16(&xs[v * 4], inputs + (size_t)b0 * I_DIM + v * 4);
  }
  wait_async0();
  __syncthreads();
  // affine input map in place: x = in*w + b
  constexpr int XFIX = (BB * I_DIM) / NTHREADS;          // = 4
#pragma unroll
  for (int k = 0; k < XFIX; ++k) {
    const int e = t + k * NTHREADS;
    const int i = e & (I_DIM - 1);
    xs[e] = __builtin_fmaf(xs[e], input_w[i], input_b[i]);
  }
  __syncthreads();

  const int n = n0 + t;
  float num[BB], den[BB];
#pragma unroll
  for (int bb = 0; bb < BB; ++bb) { num[bb] = 0.f; den[bb] = 0.f; }

#pragma unroll 2
  for (int i = 0; i < I_DIM; ++i) {
    const float4 p = psens[i * N_DIM + n];   // one b128, coalesced
#pragma unroll
    for (int bb = 0; bb < BB; ++bb) {
      const float x   = xs[bb * I_DIM + i];           // LDS broadcast
      const float arg = __builtin_fmaf(x, p.y, p.x);  // -sigma*(x-mu)*log2e
      const float s   = frcp(1.0f + fexp2(arg));      // sigmoid
      num[bb] = __builtin_fmaf(p.z, s, num[bb]);
      den[bb] = __builtin_fmaf(p.w, s, den[bb]);
    }
  }
#pragma unroll
  for (int bb = 0; bb < BB; ++bb) {
    const size_t o = (size_t)(b0 + bb) * N_DIM + n;
    wnum_s[o] = num[bb];
    wden_s[o] = den[bb];
  }
}

// ---------------------------------------------------------------------------
// Kernel 2: one ODE unfold step: vin -> vout
// ---------------------------------------------------------------------------
__global__ __launch_bounds__(NTHREADS) void ltc_step(
    const float* __restrict__ vin, float* __restrict__ vout,
    const float4* __restrict__ prec,      // packed [N,N]
    const float* __restrict__ wnum_s, const float* __restrict__ wden_s,
    const float* __restrict__ vleak, const float* __restrict__ gleak,
    const float* __restrict__ cm_t)
{
  __shared__ __align__(16) float vs[BB * N_DIM];   // 4 KB
  const int b0 = (blockIdx.x >> 1) * BB;
  const int n0 = (blockIdx.x & 1) * BN;
  const int t  = threadIdx.x;

  // async-stage the BB x 256 v_pre tile into LDS (compile-time trip count)
  constexpr int VCOPIES = (BB * N_DIM / 4) / NTHREADS;   // = 2
#pragma unroll
  for (int k = 0; k < VCOPIES; ++k) {
    const int v = t + k * NTHREADS;
    async_copy16(&vs[v * 4], vin + (size_t)b0 * N_DIM + v * 4);
  }
  wait_async0();
  __syncthreads();

  const int n = n0 + t;
  float num[BB], den[BB];
#pragma unroll
  for (int bb = 0; bb < BB; ++bb) { num[bb] = 0.f; den[bb] = 0.f; }

#pragma unroll 2
  for (int j = 0; j < N_DIM; ++j) {
    const float4 p = prec[j * N_DIM + n];   // one b128, coalesced, L2-resident
#pragma unroll
    for (int bb = 0; bb < BB; ++bb) {
      const float v   = vs[bb * N_DIM + j];           // LDS broadcast
      const float arg = __builtin_fmaf(v, p.y, p.x);
      const float s   = frcp(1.0f + fexp2(arg));
      num[bb] = __builtin_fmaf(p.z, s, num[bb]);
      den[bb] = __builtin_fmaf(p.w, s, den[bb]);
    }
  }

  const float gl = gleak[n];
  const float cm = cm_t[n];
  const float glvl = gl * vleak[n];
#pragma unroll
  for (int bb = 0; bb < BB; ++bb) {
    const size_t o = (size_t)(b0 + bb) * N_DIM + n;
    const float vp = vs[bb * N_DIM + n];
    const float numer = __builtin_fmaf(cm, vp, glvl) + num[bb] + wnum_s[o];
    const float denom = (cm + gl) + (den[bb] + wden_s[o]);
    vout[o] = numer / denom;   // once per output; keep full precision
  }
}

// ---------------------------------------------------------------------------
extern "C" void kernel_launch(void* const* d_in, const int* in_sizes, int n_in,
                              void* d_out, int out_size, void* d_ws, size_t ws_size,
                              hipStream_t stream) {
  const float* inputs  = (const float*)d_in[0];
  const float* state   = (const float*)d_in[1];
  const float* input_w = (const float*)d_in[2];
  const float* input_b = (const float*)d_in[3];
  const float* smu     = (const float*)d_in[4];
  const float* ssig    = (const float*)d_in[5];
  const float* sW      = (const float*)d_in[6];
  const float* serev   = (const float*)d_in[7];
  const float* mu      = (const float*)d_in[8];
  const float* sigma   = (const float*)d_in[9];
  const float* W       = (const float*)d_in[10];
  const float* erev    = (const float*)d_in[11];
  const float* vleak   = (const float*)d_in[12];
  const float* gleak   = (const float*)d_in[13];
  const float* cm_t    = (const float*)d_in[14];
  float* out = (float*)d_out;

  const size_t BN_ELT = (size_t)B_TOT * N_DIM;           // 262144
  float*  wnum_s = (float*)d_ws;                         // [B,N]
  float*  wden_s = wnum_s + BN_ELT;                      // [B,N]
  float*  vtmp   = wden_s + BN_ELT;                      // [B,N]
  float4* prec   = (float4*)(vtmp + BN_ELT);             // [N,N] packed (1 MB)
  float4* psens  = prec + (size_t)N_DIM * N_DIM;         // [I,N] packed (0.5 MB)

  const dim3 block(NTHREADS);                            // 128 threads (4 waves)
  const dim3 grid((B_TOT / BB) * (N_DIM / BN));          // 512 blocks

  // pack launch-invariant params (rebuilt every call; deterministic)
  ltc_pack<<<(N_DIM * N_DIM) / 256, 256, 0, stream>>>(mu, sigma, W, erev,
                                                      prec, N_DIM * N_DIM);
  ltc_pack<<<(I_DIM * N_DIM) / 256, 256, 0, stream>>>(smu, ssig, sW, serev,
                                                      psens, I_DIM * N_DIM);

  ltc_sensory<<<grid, block, 0, stream>>>(inputs, input_w, input_b,
                                          psens, wnum_s, wden_s);

  const float* src = state;
  float* buf[2] = { vtmp, out };   // parity: u=0..5 -> vtmp,out,vtmp,out,vtmp,out
  for (int u = 0; u < UNFOLDS; ++u) {
    float* dst = buf[u & 1];
    ltc_step<<<grid, block, 0, stream>>>(src, dst, prec,
                                         wnum_s, wden_s, vleak, gleak, cm_t);
    src = dst;
  }
}